// Attention_56607668961558
// MI455X (gfx1250) — compile-verified
//
#include <hip/hip_runtime.h>

typedef __attribute__((ext_vector_type(2))) float v2f;
typedef __attribute__((ext_vector_type(8))) float v8f;

#define B_ 64
#define C_ 512
#define G_ 1024
#define N_ 1024   // H*W

// ---------------------------------------------------------------------------
// K1: gbf_pj[64,512] = gbf[64,1024] @ W_proj[1024,512] with V_WMMA_F32_16X16X4_F32
// One 16x16 output tile per wave. grid = 32 blocks (N tiles), 4 waves (M tiles).
// A 16x4 f32 layout: lanes 0-15 hold M=0..15; VGPR0/VGPR1 = K{0,1} (lanes<16)
// or K{2,3} (lanes>=16).  B 4x16: VGPR0 = rows K0|K2 striped over lane halves,
// VGPR1 = rows K1|K3.  D 16x16: VGPR r, lanes 0-15 -> M=r, lanes 16-31 -> M=r+8.
// ---------------------------------------------------------------------------
__global__ __launch_bounds__(128) void wmma_proj_kernel(const float* __restrict__ gbf,
                                                        const float* __restrict__ wproj,
                                                        float* __restrict__ gbf_pj) {
    const int lane = threadIdx.x & 31;
    const int wave = threadIdx.x >> 5;   // 0..3 -> M tile
    const int half = lane >> 4;          // 0 or 1
    const int lm   = lane & 15;
    const int m0   = wave * 16;
    const int n0   = blockIdx.x * 16;

    v8f acc = {0.f, 0.f, 0.f, 0.f, 0.f, 0.f, 0.f, 0.f};

    const float* arow = gbf + (size_t)(m0 + lm) * G_;
    const float* bcol = wproj + n0 + lm;

    for (int k0 = 0; k0 < G_; k0 += 4) {
        const int ka = k0 + 2 * half;
        v2f a = *(const v2f*)(arow + ka);            // global_load_b64
        v2f b;
        b.x = bcol[(size_t)ka * C_];
        b.y = bcol[(size_t)(ka + 1) * C_];
        acc = __builtin_amdgcn_wmma_f32_16x16x4_f32(
            /*neg_a=*/false, a, /*neg_b=*/false, b,
            /*c_mod=*/(short)0, acc, /*reuse_a=*/false, /*reuse_b=*/false);
    }

#pragma unroll
    for (int r = 0; r < 8; ++r) {
        const int row = m0 + r + 8 * half;
        gbf_pj[(size_t)row * C_ + n0 + lm] = acc[r];
    }
}

// ---------------------------------------------------------------------------
// K2: bias[b] = dot(gbf_pj[b,:], W_pc)      grid = 64, block = 256
// ---------------------------------------------------------------------------
__global__ __launch_bounds__(256) void bias_kernel(const float* __restrict__ gbf_pj,
                                                   const float* __restrict__ wpc,
                                                   float* __restrict__ bias) {
    __shared__ float red[256];
    const int b = blockIdx.x, t = threadIdx.x;
    const float* row = gbf_pj + (size_t)b * C_;
    red[t] = fmaf(row[t], wpc[t], row[t + 256] * wpc[t + 256]);
    __syncthreads();
    for (int s = 128; s > 0; s >>= 1) {
        if (t < s) red[t] += red[t + s];
        __syncthreads();
    }
    if (t == 0) bias[b] = red[0];
}

// ---------------------------------------------------------------------------
// K3: c[b,n] = dot(lcf[b,n,:], W_pc) + bias[b]
// grid = (16, 64): 16 blocks of 64 rows per batch. block = 256 (8 waves),
// each wave owns 8 rows; lanes do float4 strided loads (coalesced 512B/wave)
// and reduce via __shfl_xor (wave32).
// ---------------------------------------------------------------------------
__global__ __launch_bounds__(256) void scores_kernel(const float* __restrict__ lcf,
                                                     const float* __restrict__ wpc,
                                                     const float* __restrict__ bias,
                                                     float* __restrict__ cbuf) {
    __shared__ __align__(16) float s_wpc[C_];
    const int t = threadIdx.x;
    s_wpc[t]       = wpc[t];
    s_wpc[t + 256] = wpc[t + 256];
    __syncthreads();

    const int b    = blockIdx.y;
    const int lane = t & 31;
    const int wave = t >> 5;
    const float bb = bias[b];
    const float4* w4 = (const float4*)s_wpc;

    for (int i = 0; i < 8; ++i) {
        const int n = blockIdx.x * 64 + wave * 8 + i;
        const float4* row4 = (const float4*)(lcf + ((size_t)b * N_ + n) * C_);
        float acc = 0.f;
#pragma unroll
        for (int j = 0; j < 4; ++j) {
            const float4 v = row4[lane + 32 * j];
            const float4 w = w4[lane + 32 * j];
            acc = fmaf(v.x, w.x, acc);
            acc = fmaf(v.y, w.y, acc);
            acc = fmaf(v.z, w.z, acc);
            acc = fmaf(v.w, w.w, acc);
        }
#pragma unroll
        for (int off = 16; off > 0; off >>= 1) acc += __shfl_xor(acc, off, 32);
        if (lane == 0) cbuf[(size_t)b * N_ + n] = acc + bb;
    }
}

// ---------------------------------------------------------------------------
// K4: per-batch softmax over N=1024; writes a to d_out and (in-place) to cbuf;
// also zero-inits the ga region of d_out (harness poisons it).
// grid = 64, block = 256 (4 elements/thread).
// ---------------------------------------------------------------------------
__global__ __launch_bounds__(256) void softmax_kernel(float* __restrict__ cbuf,
                                                      float* __restrict__ out_ga,
                                                      float* __restrict__ out_a) {
    __shared__ float red[256];
    const int b = blockIdx.x, t = threadIdx.x;
    float* crow = cbuf + (size_t)b * N_;

    const float v0 = crow[t], v1 = crow[t + 256], v2 = crow[t + 512], v3 = crow[t + 768];

    float m = fmaxf(fmaxf(v0, v1), fmaxf(v2, v3));
    red[t] = m;
    __syncthreads();
    for (int s = 128; s > 0; s >>= 1) {
        if (t < s) red[t] = fmaxf(red[t], red[t + s]);
        __syncthreads();
    }
    m = red[0];
    __syncthreads();

    const float e0 = __expf(v0 - m), e1 = __expf(v1 - m);
    const float e2 = __expf(v2 - m), e3 = __expf(v3 - m);
    red[t] = (e0 + e1) + (e2 + e3);
    __syncthreads();
    for (int s = 128; s > 0; s >>= 1) {
        if (t < s) red[t] += red[t + s];
        __syncthreads();
    }
    const float inv = 1.0f / red[0];

    float* arow = out_a + (size_t)b * N_;
    crow[t]       = e0 * inv;  arow[t]       = e0 * inv;
    crow[t + 256] = e1 * inv;  arow[t + 256] = e1 * inv;
    crow[t + 512] = e2 * inv;  arow[t + 512] = e2 * inv;
    crow[t + 768] = e3 * inv;  arow[t + 768] = e3 * inv;

    out_ga[(size_t)b * C_ + t]       = 0.f;
    out_ga[(size_t)b * C_ + t + 256] = 0.f;
}

// ---------------------------------------------------------------------------
// K5: ga[b,c] = sum_n a[b,n] * lcf[b,n,c]
// grid = (8, 64): each block handles 128 rows for all 512 channels.
// Thread-per-channel => every row load is a fully coalesced 2KB transaction.
// One atomicAdd per thread at the end. Second lcf pass -> L2-resident (192MB).
// ---------------------------------------------------------------------------
__global__ __launch_bounds__(512) void ga_kernel(const float* __restrict__ lcf,
                                                 const float* __restrict__ abuf,
                                                 float* __restrict__ out_ga) {
    __shared__ float s_a[128];
    const int b = blockIdx.y, t = threadIdx.x;
    const int n0 = blockIdx.x * 128;
    if (t < 128) s_a[t] = abuf[(size_t)b * N_ + n0 + t];
    __syncthreads();

    const float* base = lcf + ((size_t)b * N_ + n0) * C_ + t;
    float acc = 0.f;
#pragma unroll 8
    for (int i = 0; i < 128; ++i) acc = fmaf(s_a[i], base[(size_t)i * C_], acc);

    atomicAdd(&out_ga[(size_t)b * C_ + t], acc);
}

// ---------------------------------------------------------------------------
extern "C" void kernel_launch(void* const* d_in, const int* in_sizes, int n_in,
                              void* d_out, int out_size, void* d_ws, size_t ws_size,
                              hipStream_t stream) {
    (void)in_sizes; (void)n_in; (void)out_size; (void)ws_size;

    const float* lcf   = (const float*)d_in[0];  // [64,32,32,512]
    const float* gbf   = (const float*)d_in[1];  // [64,1024]
    const float* wproj = (const float*)d_in[2];  // [1024,512]
    const float* wpc   = (const float*)d_in[3];  // [512]

    float* out    = (float*)d_out;
    float* out_ga = out;               // [64,512]
    float* out_a  = out + B_ * C_;     // [64,1024]

    float* ws     = (float*)d_ws;
    float* gbf_pj = ws;                         // 32768 floats
    float* bias   = gbf_pj + (size_t)B_ * C_;   // 64 floats
    float* cbuf   = bias + 64;                  // 65536 floats (scores -> a)

    wmma_proj_kernel<<<dim3(C_ / 16), 128, 0, stream>>>(gbf, wproj, gbf_pj);
    bias_kernel<<<dim3(B_), 256, 0, stream>>>(gbf_pj, wpc, bias);
    scores_kernel<<<dim3(N_ / 64, B_), 256, 0, stream>>>(lcf, wpc, bias, cbuf);
    softmax_kernel<<<dim3(B_), 256, 0, stream>>>(cbuf, out_ga, out_a);
    ga_kernel<<<dim3(N_ / 128, B_), 512, 0, stream>>>(lcf, cbuf, out_ga);
}